// AttnDecoderRNN_33071248179584
// MI455X (gfx1250) — compile-verified
//
#include <hip/hip_runtime.h>
#include <math.h>

// Problem constants (match reference)
#define BB   16
#define ENC  64
#define HH   96
#define EE   128
#define VV   32000
#define KH_LIVE 47           // only conv kernel row that touches the height-1 input
#define PW   192             // padded row length: 47 zeros + 96 data + 49 zeros

typedef __attribute__((ext_vector_type(2))) float v2f;
typedef __attribute__((ext_vector_type(8))) float v8f;

// ---------------- workspace layout (floats) ----------------
#define WS_ENC_FEAT 0                      // (B,ENC,H) = 98304
#define WS_EMB      98304                  // (B,E)     = 2048
#define WS_DEC      100352                 // (B,H)     = 1536
#define WS_CVG      101888                 // (B,ENC)   = 1024
#define WS_ESC      102912                 // (B,ENC)   = 1024
#define WS_AW1      103936                 // (B,ENC)   = 1024
#define WS_COV1     104960                 // (B,ENC)   = 1024
#define WS_CTX1     105984                 // (B,H)     = 1536
#define WS_X        107520                 // (B,H)     = 1536
#define WS_GI       109056                 // (B,3H)    = 4608
#define WS_GH       113664                 // (B,3H)    = 4608
#define WS_HNEW     118272                 // (B,H)     = 1536
#define WS_OVEC     119808                 // (B,H)     = 1536
#define WS_PENC     121344                 // (B,ENC,PW)= 196608  (zero-padded enc rows)
// total ~317952 floats (~1.27 MB)

// ---------------- output layout (floats) ----------------
#define OUT_LOGP  0           // (B,V)   = 512000
#define OUT_HID   512000      // (1,B,H) = 1536
#define OUT_CTX2  513536      // (B,H)   = 1536
#define OUT_AW2   515072      // (B,ENC) = 1024
#define OUT_COV2  516096      // (B,ENC) = 1024

// ============================================================
// Kernel 0: materialize zero-padded encoder rows so the WMMA
// GEMM's B-operand loads are branchless (no EXEC predication).
//   penc[b,i,p] = (47 <= p < 143) ? enc[b,i,p-47] : 0
// ============================================================
__global__ __launch_bounds__(256) void pad_enc_kernel(
    const float* __restrict__ enc, float* __restrict__ penc)
{
  int t = blockIdx.x * 256 + threadIdx.x;      // B*ENC*PW = 196608 total
  if (t >= BB * ENC * PW) return;
  int p  = t % PW;
  int bi = t / PW;
  int hs = p - KH_LIVE;
  penc[t] = ((unsigned)hs < (unsigned)HH) ? enc[bi * HH + hs] : 0.0f;
}

// ============================================================
// Kernel 1: enc_feat = same-pad conv collapsed to a GEMM.
//   enc_feat[b,o,w] = attn_b[o]
//     + sum_i sum_kw attn_w[o,i,47,kw] * enc[b,i, w-47+kw]  (0-padded)
// GEMM: M = o (64 -> 4 tiles), N = b*96+w (1536 -> 96 tiles), K = i*96+kw.
// One wave per 16x16 tile, V_WMMA_F32_16X16X4_F32 accumulation.
// B operand comes from the padded buffer -> unconditional loads.
// ============================================================
__global__ __launch_bounds__(32) void enc_feat_wmma(
    const float* __restrict__ attn_w, const float* __restrict__ penc,
    const float* __restrict__ attn_b, float* __restrict__ enc_feat)
{
  const int lane  = threadIdx.x;            // 0..31, full wave
  const int tile  = blockIdx.x;             // 0..383
  const int o0    = (tile & 3) * 16;        // M tile
  const int n0    = (tile >> 2) * 16;       // N tile (within one b: 96 % 16 == 0)
  const int b     = n0 / HH;
  const int w0    = n0 % HH;

  const int mrow  = lane & 15;
  const int khalf = (lane < 16) ? 0 : 2;    // K split across lane halves (ISA layout)

  const int o = o0 + mrow;                  // A-matrix row for this lane
  const int w = w0 + mrow;                  // B-matrix column (output position)

  // attn_w[o][i][47][kw] ; stride per o = ENC*HH*HH
  const float* wbase = attn_w + (size_t)o * (ENC * HH * HH) + (size_t)KH_LIVE * HH;
  // padded input row base for this lane's column: index p = w + kw
  const float* ebase = penc + (size_t)b * (ENC * PW) + w;

  // C init = attn_b broadcast along rows (C VGPR r holds M = r (lanes<16) / 8+r)
  v8f c;
#pragma unroll
  for (int r = 0; r < 8; ++r) {
    int row = (lane < 16) ? r : (8 + r);
    c[r] = attn_b[o0 + row];
  }

  for (int i = 0; i < ENC; ++i) {
    const float* wi = wbase + (size_t)i * (HH * HH); // live row kh=47, contiguous
    const float* ei = ebase + (size_t)i * PW;
#pragma unroll
    for (int kw0 = 0; kw0 < HH; kw0 += 4) {
      // A: 16x4 weight tile; lane holds (M=o, K=kw0+khalf, +1) -> aligned float2
      v2f a = *(const v2f*)(wi + kw0 + khalf);
      // B: 4x16 padded-input tile; lane holds (K.., N=w) -> branchless loads
      const float* bp = ei + kw0 + khalf;
      v2f bv;
      bv.x = bp[0];
      bv.y = bp[1];
      c = __builtin_amdgcn_wmma_f32_16x16x4_f32(false, a, false, bv,
                                                (short)0, c, false, false);
    }
  }

#pragma unroll
  for (int r = 0; r < 8; ++r) {
    int row = (lane < 16) ? r : (8 + r);
    enc_feat[((size_t)b * ENC + (o0 + row)) * HH + w] = c[r];
  }
}

// ============================================================
// Kernel 2: everything tiny & serially dependent, one block.
// ============================================================
__device__ __forceinline__ void attn_step(
    const float* __restrict__ enc_feat, const float* __restrict__ enc_out,
    const float* __restrict__ hid,       // (B,H) decoder hidden
    const float* __restrict__ cov_in,    // (B,ENC)
    const float* __restrict__ W_dec, const float* __restrict__ b_dec,
    const float* __restrict__ cvg_w, const float* __restrict__ cvg_b,
    const float* __restrict__ vvec,
    float* __restrict__ dec, float* __restrict__ cvg, float* __restrict__ esc,
    float* __restrict__ aw_out, float* __restrict__ cov_out,
    float* __restrict__ ctx_out, int tid, int nt)
{
  // dec_feat[b,h] = hid[b,:] . W_dec[h,:] + b_dec[h]
  for (int t = tid; t < BB * HH; t += nt) {
    int b = t / HH, h = t % HH;
    const float* hb = hid + b * HH;
    const float* wr = W_dec + h * HH;
    float s = b_dec[h];
    for (int k = 0; k < HH; ++k) s += hb[k] * wr[k];
    dec[t] = s;
  }
  // cvg_feat[b,o] = sum_i cvg_w[o,i,0,47] * cov[b,i] + cvg_b[o]  (conv collapses to tap 47)
  for (int t = tid; t < BB * ENC; t += nt) {
    int b = t / ENC, o = t % ENC;
    float s = cvg_b[o];
    for (int i = 0; i < ENC; ++i)
      s += cvg_w[(size_t)(o * ENC + i) * HH + KH_LIVE] * cov_in[b * ENC + i];
    cvg[t] = s;
  }
  __syncthreads();
  // scores e[b,o] = sum_h tanh(enc_feat + dec + cvg) * v[b,h]
  for (int t = tid; t < BB * ENC; t += nt) {
    int b = t / ENC, o = t % ENC;
    const float* ef = enc_feat + (size_t)(b * ENC + o) * HH;
    const float* db = dec + b * HH;
    const float* vb = vvec + b * HH;
    float cv = cvg[t], s = 0.0f;
    for (int h = 0; h < HH; ++h) s += tanhf(ef[h] + db[h] + cv) * vb[h];
    esc[t] = s;
  }
  __syncthreads();
  // softmax over ENC per row + coverage update
  if (tid < BB) {
    int b = tid;
    float m = -3.4e38f;
    for (int o = 0; o < ENC; ++o) m = fmaxf(m, esc[b * ENC + o]);
    float sum = 0.0f;
    for (int o = 0; o < ENC; ++o) sum += expf(esc[b * ENC + o] - m);
    float inv = 1.0f / sum;
    for (int o = 0; o < ENC; ++o) {
      float a = expf(esc[b * ENC + o] - m) * inv;
      aw_out[b * ENC + o]  = a;
      cov_out[b * ENC + o] = cov_in[b * ENC + o] + a;
    }
  }
  __syncthreads();
  // ctx[b,h] = sum_o aw[b,o] * enc[b,o,h]
  for (int t = tid; t < BB * HH; t += nt) {
    int b = t / HH, h = t % HH;
    float s = 0.0f;
    for (int o = 0; o < ENC; ++o)
      s += aw_out[b * ENC + o] * enc_out[(size_t)(b * ENC + o) * HH + h];
    ctx_out[t] = s;
  }
  __syncthreads();
}

__global__ __launch_bounds__(1024) void middle_kernel(
    const float* __restrict__ enc_out, const int* __restrict__ input_ids,
    const float* __restrict__ hidden, const float* __restrict__ coverage,
    const float* __restrict__ emb,
    const float* __restrict__ W_dec, const float* __restrict__ b_dec,
    const float* __restrict__ cvg_w, const float* __restrict__ cvg_b,
    const float* __restrict__ vvec,
    const float* __restrict__ W_new, const float* __restrict__ b_new,
    const float* __restrict__ w_ih, const float* __restrict__ w_hh,
    const float* __restrict__ b_ih, const float* __restrict__ b_hh,
    const float* __restrict__ W_pre, const float* __restrict__ b_pre,
    float* __restrict__ ws, float* __restrict__ out)
{
  const int tid = threadIdx.x;
  const int nt  = blockDim.x;

  float* EMB  = ws + WS_EMB;
  float* DEC  = ws + WS_DEC;
  float* CVG  = ws + WS_CVG;
  float* ESC  = ws + WS_ESC;
  float* AW1  = ws + WS_AW1;
  float* COV1 = ws + WS_COV1;
  float* CTX1 = ws + WS_CTX1;
  float* XB   = ws + WS_X;
  float* GI   = ws + WS_GI;
  float* GH   = ws + WS_GH;
  float* HN   = ws + WS_HNEW;
  float* OV   = ws + WS_OVEC;
  const float* EF = ws + WS_ENC_FEAT;

  // embedding gather
  for (int t = tid; t < BB * EE; t += nt) {
    int b = t / EE, e = t % EE;
    EMB[t] = emb[(size_t)input_ids[b] * EE + e];
  }
  __syncthreads();

  // ---- attention pass 1 (hidden[0], coverage) ----
  attn_step(EF, enc_out, hidden, coverage, W_dec, b_dec, cvg_w, cvg_b, vvec,
            DEC, CVG, ESC, AW1, COV1, CTX1, tid, nt);

  // x = [emb, ctx1] @ W_new.T + b_new    (W_new: (H, E+H))
  for (int t = tid; t < BB * HH; t += nt) {
    int b = t / HH, h = t % HH;
    const float* wr = W_new + (size_t)h * (EE + HH);
    float s = b_new[h];
    const float* eb = EMB + b * EE;
    const float* cb = CTX1 + b * HH;
    for (int j = 0; j < EE; ++j) s += eb[j] * wr[j];
    for (int j = 0; j < HH; ++j) s += cb[j] * wr[EE + j];
    XB[t] = s;
  }
  __syncthreads();

  // GRU gates (PyTorch order r,z,n)
  for (int t = tid; t < BB * 3 * HH; t += nt) {
    int b = t / (3 * HH), g = t % (3 * HH);
    const float* wi = w_ih + (size_t)g * HH;
    const float* wh = w_hh + (size_t)g * HH;
    const float* xb = XB + b * HH;
    const float* hb = hidden + b * HH;
    float si = b_ih[g], sh = b_hh[g];
    for (int k = 0; k < HH; ++k) { si += xb[k] * wi[k]; sh += hb[k] * wh[k]; }
    GI[t] = si; GH[t] = sh;
  }
  __syncthreads();
  for (int t = tid; t < BB * HH; t += nt) {
    int b = t / HH, h = t % HH;
    float r = 1.0f / (1.0f + expf(-(GI[b*3*HH + h]          + GH[b*3*HH + h])));
    float z = 1.0f / (1.0f + expf(-(GI[b*3*HH + HH + h]     + GH[b*3*HH + HH + h])));
    float n = tanhf(GI[b*3*HH + 2*HH + h] + r * GH[b*3*HH + 2*HH + h]);
    float hv = hidden[t];
    float hn = (1.0f - z) * n + z * hv;
    HN[t] = hn;
    out[OUT_HID + t] = hn;                 // hidden_new (1,B,H)
  }
  __syncthreads();

  // ---- attention pass 2 (h_new, cov1) -> outputs ----
  attn_step(EF, enc_out, HN, COV1, W_dec, b_dec, cvg_w, cvg_b, vvec,
            DEC, CVG, ESC, out + OUT_AW2, out + OUT_COV2, out + OUT_CTX2, tid, nt);

  // o = tanh([h_new, ctx2] @ W_pre.T + b_pre)
  for (int t = tid; t < BB * HH; t += nt) {
    int b = t / HH, h = t % HH;
    const float* wr = W_pre + (size_t)h * (2 * HH);
    const float* hb = HN + b * HH;
    const float* cb = out + OUT_CTX2 + b * HH;
    float s = b_pre[h];
    for (int k = 0; k < HH; ++k) s += hb[k] * wr[k] + cb[k] * wr[HH + k];
    OV[t] = tanhf(s);
  }
}

// ============================================================
// Kernel 3: logits[b,n] = o[b,:] . W_out[n,:] + b_out[n]
// M = B = 16 (one tile), N = 32000 (2000 tiles), K = 96 (24 x K=4).
// ============================================================
__global__ __launch_bounds__(32) void logits_wmma(
    const float* __restrict__ ovec, const float* __restrict__ W_out,
    const float* __restrict__ b_out, float* __restrict__ logits)
{
  const int lane  = threadIdx.x;
  const int n0    = blockIdx.x * 16;
  const int col   = n0 + (lane & 15);
  const int arow  = lane & 15;
  const int khalf = (lane < 16) ? 0 : 2;

  v8f c;
  float bias = b_out[col];
#pragma unroll
  for (int r = 0; r < 8; ++r) c[r] = bias;

#pragma unroll
  for (int k0 = 0; k0 < HH; k0 += 4) {
    v2f a = *(const v2f*)(ovec + (size_t)arow * HH + k0 + khalf);
    v2f b = *(const v2f*)(W_out + (size_t)col * HH + k0 + khalf);
    c = __builtin_amdgcn_wmma_f32_16x16x4_f32(false, a, false, b,
                                              (short)0, c, false, false);
  }

#pragma unroll
  for (int r = 0; r < 8; ++r) {
    int row = (lane < 16) ? r : (8 + r);
    logits[(size_t)row * VV + col] = c[r];
  }
}

// ============================================================
// Kernel 4: in-place log_softmax over V per batch row.
// ============================================================
__global__ __launch_bounds__(256) void logsoftmax_kernel(float* __restrict__ logp)
{
  __shared__ float red[256];
  const int b = blockIdx.x, tid = threadIdx.x;
  float* row = logp + (size_t)b * VV;

  float m = -3.4e38f;
  for (int i = tid; i < VV; i += 256) m = fmaxf(m, row[i]);
  red[tid] = m; __syncthreads();
  for (int s = 128; s > 0; s >>= 1) { if (tid < s) red[tid] = fmaxf(red[tid], red[tid+s]); __syncthreads(); }
  m = red[0]; __syncthreads();

  float sum = 0.0f;
  for (int i = tid; i < VV; i += 256) sum += expf(row[i] - m);
  red[tid] = sum; __syncthreads();
  for (int s = 128; s > 0; s >>= 1) { if (tid < s) red[tid] += red[tid+s]; __syncthreads(); }
  const float lse = m + logf(red[0]);
  __syncthreads();

  for (int i = tid; i < VV; i += 256) row[i] = row[i] - lse;
}

extern "C" void kernel_launch(void* const* d_in, const int* in_sizes, int n_in,
                              void* d_out, int out_size, void* d_ws, size_t ws_size,
                              hipStream_t stream)
{
  const float* enc_out   = (const float*)d_in[0];
  const int*   input_ids = (const int*)  d_in[1];
  const float* hidden    = (const float*)d_in[2];
  const float* coverage  = (const float*)d_in[3];
  const float* emb       = (const float*)d_in[4];
  const float* W_dec     = (const float*)d_in[5];
  const float* b_dec     = (const float*)d_in[6];
  const float* attn_w    = (const float*)d_in[7];
  const float* attn_b    = (const float*)d_in[8];
  const float* cvg_w     = (const float*)d_in[9];
  const float* cvg_b     = (const float*)d_in[10];
  const float* vvec      = (const float*)d_in[11];
  const float* W_new     = (const float*)d_in[12];
  const float* b_new     = (const float*)d_in[13];
  const float* w_ih      = (const float*)d_in[14];
  const float* w_hh      = (const float*)d_in[15];
  const float* b_ih      = (const float*)d_in[16];
  const float* b_hh      = (const float*)d_in[17];
  const float* W_pre     = (const float*)d_in[18];
  const float* b_pre     = (const float*)d_in[19];
  const float* W_out     = (const float*)d_in[20];
  const float* b_out     = (const float*)d_in[21];

  float* out = (float*)d_out;
  float* ws  = (float*)d_ws;

  // 0) zero-padded encoder rows (makes the WMMA B-loads branchless)
  pad_enc_kernel<<<(BB * ENC * PW + 255) / 256, 256, 0, stream>>>(enc_out, ws + WS_PENC);

  // 1) conv collapsed to WMMA-f32 GEMM (reads only the live kh=47 slice: 1.6 MB of 151 MB)
  enc_feat_wmma<<<384, 32, 0, stream>>>(attn_w, ws + WS_PENC, attn_b, ws + WS_ENC_FEAT);

  // 2) all serial small stages in one block
  middle_kernel<<<1, 1024, 0, stream>>>(enc_out, input_ids, hidden, coverage, emb,
                                        W_dec, b_dec, cvg_w, cvg_b, vvec,
                                        W_new, b_new, w_ih, w_hh, b_ih, b_hh,
                                        W_pre, b_pre, ws, out);

  // 3) vocab projection (BW-bound on W_out: 12.3 MB)
  logits_wmma<<<VV / 16, 32, 0, stream>>>(ws + WS_OVEC, W_out, b_out, out + OUT_LOGP);

  // 4) in-place log_softmax
  logsoftmax_kernel<<<BB, 256, 0, stream>>>(out + OUT_LOGP);
}